// GATEncoder_7421703487981
// MI455X (gfx1250) — compile-verified
//
#include <hip/hip_runtime.h>

typedef __attribute__((ext_vector_type(2))) float v2f;
typedef __attribute__((ext_vector_type(8))) float v8f;

#define NN       50000
#define IN_CH    128
#define OUT_CH   64
#define HEADS    2
#define HID      128                 // 2*OUT_CH per head in conv1
#define HC1      (HEADS*HID)         // 256
#define E0_EDGES 400000
#define E_TOT    (E0_EDGES + NN)     // 450000 incl. self loops
#define NEG_SLOPE 0.2f

// ---------------------------------------------------------------------------
// WMMA fp32 GEMM: C[M,N] = A[M,K] @ B[K,N], all row-major.
// Requires M%16==0, N%64==0, K%4==0. One wave -> 16x64 tile (4 wmma accums),
// software-pipelined K loop (next K-step loads overlap current WMMAs).
// Operand layouts per CDNA5 ISA 7.12.2 (f32 16x16x4):
//   A/B vector elem j in lane l holds K = j + 2*(l/16)
//   C/D elem j in lane l is row j + 8*(l/16), col l%16
// ---------------------------------------------------------------------------
struct Bop { v2f b0, b1, b2, b3; };

__device__ __forceinline__ v2f load_a(const float* __restrict__ arow, int k0, int lhalf) {
  v2f a;
  a.x = arow[k0 + 2 * lhalf];
  a.y = arow[k0 + 2 * lhalf + 1];
  return a;
}
__device__ __forceinline__ Bop load_b(const float* __restrict__ B, int k0, int N,
                                      int tn, int lhalf, int lmod) {
  const float* r0 = B + (size_t)(k0 + 2 * lhalf) * N + tn + lmod;
  const float* r1 = r0 + N;
  Bop b;
  b.b0.x = r0[0];  b.b0.y = r1[0];
  b.b1.x = r0[16]; b.b1.y = r1[16];
  b.b2.x = r0[32]; b.b2.y = r1[32];
  b.b3.x = r0[48]; b.b3.y = r1[48];
  return b;
}

__global__ void __launch_bounds__(256)
gemm_wmma_f32(const float* __restrict__ A, const float* __restrict__ B,
              float* __restrict__ C, int M, int N, int K) {
  const int wave   = blockIdx.x * (blockDim.x >> 5) + (threadIdx.x >> 5);
  const int tilesN = N >> 6;
  const int tm     = wave / tilesN;
  const int tn     = (wave % tilesN) << 6;
  if (tm * 16 >= M) return;                    // uniform per wave
  const int lhalf = (threadIdx.x >> 4) & 1;
  const int lmod  = threadIdx.x & 15;

  v8f c0 = {}, c1 = {}, c2 = {}, c3 = {};
  const float* arow = A + (size_t)(tm * 16 + lmod) * K;

  v2f a = load_a(arow, 0, lhalf);
  Bop b = load_b(B, 0, N, tn, lhalf, lmod);
  for (int k0 = 0; k0 < K - 4; k0 += 4) {
    v2f an = load_a(arow, k0 + 4, lhalf);           // prefetch next K-step
    Bop bn = load_b(B, k0 + 4, N, tn, lhalf, lmod);
    c0 = __builtin_amdgcn_wmma_f32_16x16x4_f32(false, a, false, b.b0, (short)0, c0, false, false);
    c1 = __builtin_amdgcn_wmma_f32_16x16x4_f32(false, a, false, b.b1, (short)0, c1, false, false);
    c2 = __builtin_amdgcn_wmma_f32_16x16x4_f32(false, a, false, b.b2, (short)0, c2, false, false);
    c3 = __builtin_amdgcn_wmma_f32_16x16x4_f32(false, a, false, b.b3, (short)0, c3, false, false);
    a = an; b = bn;
  }
  c0 = __builtin_amdgcn_wmma_f32_16x16x4_f32(false, a, false, b.b0, (short)0, c0, false, false);
  c1 = __builtin_amdgcn_wmma_f32_16x16x4_f32(false, a, false, b.b1, (short)0, c1, false, false);
  c2 = __builtin_amdgcn_wmma_f32_16x16x4_f32(false, a, false, b.b2, (short)0, c2, false, false);
  c3 = __builtin_amdgcn_wmma_f32_16x16x4_f32(false, a, false, b.b3, (short)0, c3, false, false);

#pragma unroll
  for (int j = 0; j < 8; ++j) {
    float* p = C + (size_t)(tm * 16 + j + 8 * lhalf) * N + tn + lmod;
    p[0] = c0[j]; p[16] = c1[j]; p[32] = c2[j]; p[48] = c3[j];
  }
}

// a_src[n,h] = h[n,h,:]·att_s[h,:]   a_dst likewise  (C % 4 == 0)
__global__ void __launch_bounds__(256)
att_scores(const float* __restrict__ h, const float* __restrict__ att_s,
           const float* __restrict__ att_d, float* __restrict__ as,
           float* __restrict__ ad, int n_nodes, int H, int C) {
  int i = blockIdx.x * blockDim.x + threadIdx.x;
  if (i >= n_nodes * H) return;
  int n = i / H, hh = i % H;
  const float4* hp = (const float4*)(h + (size_t)n * H * C + (size_t)hh * C);
  const float4* sp = (const float4*)(att_s + hh * C);
  const float4* dp = (const float4*)(att_d + hh * C);
  float s = 0.f, d = 0.f;
  for (int c = 0; c < (C >> 2); ++c) {
    float4 v = hp[c], ws = sp[c], wd = dp[c];
    s += v.x * ws.x + v.y * ws.y + v.z * ws.z + v.w * ws.w;
    d += v.x * wd.x + v.y * wd.y + v.z * wd.z + v.w * wd.w;
  }
  as[i] = s; ad[i] = d;
}

__device__ __forceinline__ void edge_nodes(const int* __restrict__ ei, int e,
                                           int& s, int& d) {
  if (e < E0_EDGES) { s = ei[e]; d = ei[E0_EDGES + e]; }
  else              { s = e - E0_EDGES; d = s; }            // self loop
}
__device__ __forceinline__ float leaky(float v) { return v >= 0.f ? v : NEG_SLOPE * v; }
// monotonic float<->uint order encoding (max over it == float max)
__device__ __forceinline__ unsigned f2ord(float f) {
  unsigned u = __float_as_uint(f);
  return (u & 0x80000000u) ? ~u : (u | 0x80000000u);
}
__device__ __forceinline__ float ord2f(unsigned k) {
  unsigned u = (k & 0x80000000u) ? (k ^ 0x80000000u) : ~k;
  return __uint_as_float(u);
}

__global__ void __launch_bounds__(256)
edge_max(const int* __restrict__ ei, const float* __restrict__ as,
         const float* __restrict__ ad, unsigned* __restrict__ m, int H) {
  int i = blockIdx.x * blockDim.x + threadIdx.x;
  if (i >= E_TOT * H) return;
  int e = i / H, hh = i % H;
  int s, d; edge_nodes(ei, e, s, d);
  float v = leaky(as[s * H + hh] + ad[d * H + hh]);
  atomicMax(&m[d * H + hh], f2ord(v));
}

__global__ void __launch_bounds__(256)
edge_exp(const int* __restrict__ ei, const float* __restrict__ as,
         const float* __restrict__ ad, const unsigned* __restrict__ m,
         float* __restrict__ ea, float* __restrict__ denom, int H) {
  int i = blockIdx.x * blockDim.x + threadIdx.x;
  if (i >= E_TOT * H) return;
  int e = i / H, hh = i % H;
  int s, d; edge_nodes(ei, e, s, d);
  float v  = leaky(as[s * H + hh] + ad[d * H + hh]);
  float ex = __expf(v - ord2f(m[d * H + hh]));
  ea[i] = ex;
  atomicAdd(&denom[d * H + hh], ex);
}

// agg[dst,:] += alpha * h[src,:], 4 channels per thread
__global__ void __launch_bounds__(256)
edge_scatter(const int* __restrict__ ei, const float* __restrict__ h,
             const float* __restrict__ ea, const float* __restrict__ denom,
             float* __restrict__ agg, int H, int C) {
  const int HC  = H * C;
  const int per = HC >> 2;
  long long i = (long long)blockIdx.x * blockDim.x + threadIdx.x;
  if (i >= (long long)E_TOT * per) return;
  int e  = (int)(i / per);
  int cq = (int)(i % per) << 2;
  int hh = cq / C;
  int s, d; edge_nodes(ei, e, s, d);
  float a = ea[(size_t)e * H + hh] / (denom[d * H + hh] + 1e-16f);
  float4 hv = *(const float4*)(h + (size_t)s * HC + cq);
  float* p = agg + (size_t)d * HC + cq;
  atomicAdd(p + 0, hv.x * a);
  atomicAdd(p + 1, hv.y * a);
  atomicAdd(p + 2, hv.z * a);
  atomicAdd(p + 3, hv.w * a);
}

__global__ void __launch_bounds__(256)
bias_elu(float* __restrict__ h, const float* __restrict__ b, int total, int C) {
  int i = blockIdx.x * blockDim.x + threadIdx.x;
  if (i >= total) return;
  float v = h[i] + b[i % C];
  h[i] = v > 0.f ? v : (__expf(v) - 1.f);
}

__global__ void __launch_bounds__(256)
bias_add(float* __restrict__ o, const float* __restrict__ b, int total, int C) {
  int i = blockIdx.x * blockDim.x + threadIdx.x;
  if (i >= total) return;
  o[i] += b[i % C];
}

// ---------------------------------------------------------------------------
extern "C" void kernel_launch(void* const* d_in, const int* in_sizes, int n_in,
                              void* d_out, int out_size, void* d_ws, size_t ws_size,
                              hipStream_t stream) {
  const float* x    = (const float*)d_in[0];
  const int*   ei   = (const int*)  d_in[1];
  const float* W1   = (const float*)d_in[2];
  const float* aS1  = (const float*)d_in[3];
  const float* aD1  = (const float*)d_in[4];
  const float* b1   = (const float*)d_in[5];
  const float* Wmu  = (const float*)d_in[6];
  const float* aSmu = (const float*)d_in[7];
  const float* aDmu = (const float*)d_in[8];
  const float* bmu  = (const float*)d_in[9];
  const float* Wls  = (const float*)d_in[10];
  const float* aSls = (const float*)d_in[11];
  const float* aDls = (const float*)d_in[12];
  const float* bls  = (const float*)d_in[13];

  char* ws = (char*)d_ws;
  size_t off = 0;
  auto alloc = [&](size_t bytes) -> char* {
    off = (off + 255) & ~(size_t)255;
    char* p = ws + off; off += bytes; return p;
  };

  float*    h1   = (float*)   alloc((size_t)NN * HC1 * 4);   // conv1 features (reused for hm/hl)
  float*    hbuf = (float*)   alloc((size_t)NN * HC1 * 4);   // agg1 -> elu -> h
  float*    as1  = (float*)   alloc((size_t)NN * HEADS * 4);
  float*    ad1  = (float*)   alloc((size_t)NN * HEADS * 4);
  unsigned* m1   = (unsigned*)alloc((size_t)NN * HEADS * 4);
  float*    dn1  = (float*)   alloc((size_t)NN * HEADS * 4);
  float*    ea1  = (float*)   alloc((size_t)E_TOT * HEADS * 4);
  float*    as2m = (float*)   alloc((size_t)NN * 4);
  float*    ad2m = (float*)   alloc((size_t)NN * 4);
  unsigned* m2m  = (unsigned*)alloc((size_t)NN * 4);
  float*    dn2m = (float*)   alloc((size_t)NN * 4);
  float*    ea2m = (float*)   alloc((size_t)E_TOT * 4);
  float*    as2l = (float*)   alloc((size_t)NN * 4);
  float*    ad2l = (float*)   alloc((size_t)NN * 4);
  unsigned* m2l  = (unsigned*)alloc((size_t)NN * 4);
  float*    dn2l = (float*)   alloc((size_t)NN * 4);
  float*    ea2l = (float*)   alloc((size_t)E_TOT * 4);

  float* hm = h1;                          // NN x 64, reuse h1 space after layer 1
  float* hl = h1 + (size_t)NN * OUT_CH;
  float* out_mu = (float*)d_out;
  float* out_ls = out_mu + (size_t)NN * OUT_CH;

  // ---- layer 1 ----
  hipMemsetAsync(hbuf, 0, (size_t)NN * HC1 * 4, stream);
  hipMemsetAsync(m1,   0, (size_t)NN * HEADS * 4, stream);
  hipMemsetAsync(dn1,  0, (size_t)NN * HEADS * 4, stream);

  int waves = (NN / 16) * (HC1 / 64);                       // 12500
  gemm_wmma_f32<<<(waves + 7) / 8, 256, 0, stream>>>(x, W1, h1, NN, HC1, IN_CH);
  att_scores<<<(NN * HEADS + 255) / 256, 256, 0, stream>>>(h1, aS1, aD1, as1, ad1, NN, HEADS, HID);
  edge_max<<<(E_TOT * HEADS + 255) / 256, 256, 0, stream>>>(ei, as1, ad1, m1, HEADS);
  edge_exp<<<(E_TOT * HEADS + 255) / 256, 256, 0, stream>>>(ei, as1, ad1, m1, ea1, dn1, HEADS);
  {
    long long work = (long long)E_TOT * (HC1 / 4);
    edge_scatter<<<(unsigned)((work + 255) / 256), 256, 0, stream>>>(ei, h1, ea1, dn1, hbuf, HEADS, HID);
  }
  bias_elu<<<(NN * HC1 + 255) / 256, 256, 0, stream>>>(hbuf, b1, NN * HC1, HC1);

  // ---- layer 2 GEMMs (h1 region now free) ----
  waves = (NN / 16) * (OUT_CH / 64);                        // 3125
  gemm_wmma_f32<<<(waves + 7) / 8, 256, 0, stream>>>(hbuf, Wmu, hm, NN, OUT_CH, HC1);
  gemm_wmma_f32<<<(waves + 7) / 8, 256, 0, stream>>>(hbuf, Wls, hl, NN, OUT_CH, HC1);
  att_scores<<<(NN + 255) / 256, 256, 0, stream>>>(hm, aSmu, aDmu, as2m, ad2m, NN, 1, OUT_CH);
  att_scores<<<(NN + 255) / 256, 256, 0, stream>>>(hl, aSls, aDls, as2l, ad2l, NN, 1, OUT_CH);

  hipMemsetAsync(m2m, 0, (size_t)NN * 4, stream);
  hipMemsetAsync(dn2m, 0, (size_t)NN * 4, stream);
  hipMemsetAsync(m2l, 0, (size_t)NN * 4, stream);
  hipMemsetAsync(dn2l, 0, (size_t)NN * 4, stream);
  hipMemsetAsync(d_out, 0, (size_t)2 * NN * OUT_CH * 4, stream);

  // ---- conv_mu ----
  edge_max<<<(E_TOT + 255) / 256, 256, 0, stream>>>(ei, as2m, ad2m, m2m, 1);
  edge_exp<<<(E_TOT + 255) / 256, 256, 0, stream>>>(ei, as2m, ad2m, m2m, ea2m, dn2m, 1);
  {
    long long work = (long long)E_TOT * (OUT_CH / 4);
    edge_scatter<<<(unsigned)((work + 255) / 256), 256, 0, stream>>>(ei, hm, ea2m, dn2m, out_mu, 1, OUT_CH);
  }
  // ---- conv_logstd ----
  edge_max<<<(E_TOT + 255) / 256, 256, 0, stream>>>(ei, as2l, ad2l, m2l, 1);
  edge_exp<<<(E_TOT + 255) / 256, 256, 0, stream>>>(ei, as2l, ad2l, m2l, ea2l, dn2l, 1);
  {
    long long work = (long long)E_TOT * (OUT_CH / 4);
    edge_scatter<<<(unsigned)((work + 255) / 256), 256, 0, stream>>>(ei, hl, ea2l, dn2l, out_ls, 1, OUT_CH);
  }

  bias_add<<<(NN * OUT_CH + 255) / 256, 256, 0, stream>>>(out_mu, bmu, NN * OUT_CH, OUT_CH);
  bias_add<<<(NN * OUT_CH + 255) / 256, 256, 0, stream>>>(out_ls, bls, NN * OUT_CH, OUT_CH);
}